// GCN_71949292143070
// MI455X (gfx1250) — compile-verified
//
#include <hip/hip_runtime.h>
#include <hip/hip_bf16.h>
#include <stdint.h>
#include <stddef.h>

typedef float v2f __attribute__((ext_vector_type(2)));
typedef float v8f __attribute__((ext_vector_type(8)));

#define NN 65536
#define NE 524288
#define NG 64
#define PTS 1024

// ---------------- degree / normalization ----------------
__global__ void k_init_deg(float* deg, int n) {
    int i = blockIdx.x * blockDim.x + threadIdx.x;
    if (i < n) deg[i] = 1.0f;  // self-loop
}

__global__ void k_edge_deg(const int* __restrict__ col, float* deg, int e) {
    int i = blockIdx.x * blockDim.x + threadIdx.x;
    if (i < e) atomicAdd(&deg[col[i]], 1.0f);
}

__global__ void k_dinv(float* d, int n) {
    int i = blockIdx.x * blockDim.x + threadIdx.x;
    if (i < n) d[i] = rsqrtf(d[i]);
}

// ---------------- layer-1 GEMM (K=3, too thin for WMMA) ----------------
__global__ void k_gemm_k3(const float* __restrict__ X, const float* __restrict__ W,
                          float* __restrict__ Y, int n) {
    int i = blockIdx.x * blockDim.x + threadIdx.x;  // n*64 threads
    if (i < n * 64) {
        int r = i >> 6, f = i & 63;
        const float* xr = X + r * 3;
        Y[i] = xr[0] * W[f] + xr[1] * W[64 + f] + xr[2] * W[128 + f];
    }
}

// ---------------- WMMA GEMM: Y[N,F] = X[N,K] @ W[K,F] (+bias, +relu) ----
// Register-blocked: each wave owns a 64x16 output tile (4 accumulators,
// M-blocked) so the strided B (weight) fragment is loaded ONCE per k-step
// and reused by 4 V_WMMA_F32_16X16X4_F32 ops. 4 waves/block cover 64 cols.
// Per k-step per wave: 4 b64 A-loads + 2 b32 B-loads -> 4 wmma (8192 FLOP).
__global__ void __launch_bounds__(128)
k_gemm_wmma(const float* __restrict__ X, const float* __restrict__ W,
            float* __restrict__ Y, int K, int F,
            const float* __restrict__ bias, int do_relu) {
    const int lane = threadIdx.x & 31;
    const int wave = threadIdx.x >> 5;
    const int mbase = blockIdx.y << 6;                 // 64 rows per block
    const int n0    = (blockIdx.x * 4 + wave) << 4;    // 16 cols per wave
    const int half  = lane >> 4;   // selects K pair {0,1} vs {2,3}
    const int lm    = lane & 15;   // M index for A, N index for B

    v8f acc0 = {}, acc1 = {}, acc2 = {}, acc3 = {};

    // A fragments: rows (mbase + j*16 + lm), columns k+2*half, k+2*half+1
    const float* xr = X + (size_t)(mbase + lm) * K + 2 * half;
    const size_t rstep = (size_t)16 * K;               // 16 rows down
    // B fragment: rows k+2*half, k+2*half+1, column (n0+lm)
    const float* wc = W + (size_t)(2 * half) * F + (n0 + lm);

    for (int k = 0; k < K; k += 4) {
        v2f b;
        b.x = wc[0];
        b.y = wc[F];
        v2f a0; a0.x = xr[0];             a0.y = xr[1];
        v2f a1; a1.x = xr[rstep];         a1.y = xr[rstep + 1];
        v2f a2; a2.x = xr[2 * rstep];     a2.y = xr[2 * rstep + 1];
        v2f a3; a3.x = xr[3 * rstep];     a3.y = xr[3 * rstep + 1];
        acc0 = __builtin_amdgcn_wmma_f32_16x16x4_f32(false, a0, false, b, (short)0, acc0, false, false);
        acc1 = __builtin_amdgcn_wmma_f32_16x16x4_f32(false, a1, false, b, (short)0, acc1, false, false);
        acc2 = __builtin_amdgcn_wmma_f32_16x16x4_f32(false, a2, false, b, (short)0, acc2, false, false);
        acc3 = __builtin_amdgcn_wmma_f32_16x16x4_f32(false, a3, false, b, (short)0, acc3, false, false);
        xr += 4;
        wc += (size_t)4 * F;
    }

    const int c = n0 + lm;
    const float bv = bias ? bias[c] : 0.0f;
    // D layout: VGPR g holds row m0+g (lanes 0-15) / m0+8+g (lanes 16-31)
#pragma unroll
    for (int g = 0; g < 8; ++g) {
        int r = mbase + g + 8 * half;
        float v0 = acc0[g] + bv;
        float v1 = acc1[g] + bv;
        float v2 = acc2[g] + bv;
        float v3 = acc3[g] + bv;
        if (do_relu) {
            v0 = fmaxf(v0, 0.0f); v1 = fmaxf(v1, 0.0f);
            v2 = fmaxf(v2, 0.0f); v3 = fmaxf(v3, 0.0f);
        }
        Y[(size_t)(r)      * F + c] = v0;
        Y[(size_t)(r + 16) * F + c] = v1;
        Y[(size_t)(r + 32) * F + c] = v2;
        Y[(size_t)(r + 48) * F + c] = v3;
    }
}

// ---------------- self-loop init: out = dinv^2 * xw ----------------
__global__ void k_selfloop(const float* __restrict__ xw, const float* __restrict__ dinv,
                           float* __restrict__ out, int n, int F) {
    long long i = (long long)blockIdx.x * blockDim.x + threadIdx.x;
    if (i < (long long)n * F) {
        int r = (int)(i / F);
        float d = dinv[r];
        out[i] = d * d * xw[i];
    }
}

// ---------------- edge scatter: out[col] += dinv[row]*dinv[col]*xw[row] ----
__global__ void k_scatter(const int* __restrict__ row, const int* __restrict__ col,
                          const float* __restrict__ xw, const float* __restrict__ dinv,
                          float* __restrict__ out, int e, int F) {
    const int fq = F >> 2;  // float4 chunks per row
    long long tid = (long long)blockIdx.x * blockDim.x + threadIdx.x;
    if (tid >= (long long)e * fq) return;
    int ed = (int)(tid / fq);
    int q  = (int)(tid % fq) << 2;
    int r = row[ed], c = col[ed];
    float nm = dinv[r] * dinv[c];
    const float4 v = *(const float4*)(xw + (size_t)r * F + q);
    float* dst = out + (size_t)c * F + q;
    atomicAdd(dst + 0, nm * v.x);
    atomicAdd(dst + 1, nm * v.y);
    atomicAdd(dst + 2, nm * v.z);
    atomicAdd(dst + 3, nm * v.w);
}

// ---------------- bias + relu ----------------
__global__ void k_bias_relu(float* __restrict__ out, const float* __restrict__ b,
                            int n, int F) {
    long long i = (long long)blockIdx.x * blockDim.x + threadIdx.x;
    if (i < (long long)n * F) {
        int f = (int)(i % F);
        out[i] = fmaxf(out[i] + b[f], 0.0f);
    }
}

// ---------------- per-graph mean/max pooling ----------------
__global__ void k_pool(const float* __restrict__ x3, float* __restrict__ h) {
    int g = blockIdx.x;            // 64 graphs
    int t = threadIdx.x;           // 256 threads, 2 features each
#pragma unroll
    for (int j = 0; j < 2; ++j) {
        int f = t + j * 256;
        const float* p = x3 + (size_t)g * PTS * 512 + f;
        float s = 0.0f, m = -3.4e38f;
        for (int i = 0; i < PTS; ++i) {
            float v = p[(size_t)i * 512];
            s += v;
            m = fmaxf(m, v);
        }
        h[g * 1024 + f]       = s * (1.0f / (float)PTS);
        h[g * 1024 + 512 + f] = m;
    }
}

// ---------------- final 512->10 layer ----------------
__global__ void k_mlp2(const float* __restrict__ h2, const float* __restrict__ Wm2,
                       const float* __restrict__ bm2, float* __restrict__ out) {
    int i = blockIdx.x * blockDim.x + threadIdx.x;  // 64*10
    if (i < NG * 10) {
        int g = i / 10, o = i % 10;
        const float* hr = h2 + g * 512;
        float s = bm2[o];
        for (int k = 0; k < 512; ++k) s += hr[k] * Wm2[k * 10 + o];
        out[i] = s;
    }
}

static inline int cdiv(long long a, int b) { return (int)((a + b - 1) / b); }

extern "C" void kernel_launch(void* const* d_in, const int* in_sizes, int n_in,
                              void* d_out, int out_size, void* d_ws, size_t ws_size,
                              hipStream_t stream) {
    const float* x   = (const float*)d_in[0];
    const int*   ei  = (const int*)d_in[1];      // edge_index [2, E] flat
    // d_in[2] = batch (node>>10, recomputed implicitly in pooling)
    const float* W1  = (const float*)d_in[3];
    const float* b1  = (const float*)d_in[4];
    const float* W2  = (const float*)d_in[5];
    const float* b2  = (const float*)d_in[6];
    const float* W3  = (const float*)d_in[7];
    const float* b3  = (const float*)d_in[8];
    const float* Wm1 = (const float*)d_in[9];
    const float* bm1 = (const float*)d_in[10];
    const float* Wm2 = (const float*)d_in[11];
    const float* bm2 = (const float*)d_in[12];
    float* out = (float*)d_out;

    const int* erow = ei;
    const int* ecol = ei + NE;

    // workspace partition (floats)
    float* ws = (float*)d_ws;
    float* dinv = ws;                      size_t o = NN;
    float* xw   = ws + o;                  o += (size_t)NN * 512;   // shared GEMM out buf
    float* x1   = ws + o;                  o += (size_t)NN * 64;
    float* x2   = ws + o;                  o += (size_t)NN * 256;
    float* x3   = ws + o;                  o += (size_t)NN * 512;
    float* h    = ws + o;                  o += (size_t)NG * 1024;
    float* h2   = ws + o;                  o += (size_t)NG * 512;

    const int B = 256;

    // normalization coefficients (layer-invariant)
    k_init_deg<<<cdiv(NN, B), B, 0, stream>>>(dinv, NN);
    k_edge_deg<<<cdiv(NE, B), B, 0, stream>>>(ecol, dinv, NE);
    k_dinv<<<cdiv(NN, B), B, 0, stream>>>(dinv, NN);

    // ---- layer 1: K=3 -> F=64 ----
    k_gemm_k3<<<cdiv((long long)NN * 64, B), B, 0, stream>>>(x, W1, xw, NN);
    k_selfloop<<<cdiv((long long)NN * 64, B), B, 0, stream>>>(xw, dinv, x1, NN, 64);
    k_scatter<<<cdiv((long long)NE * 16, B), B, 0, stream>>>(erow, ecol, xw, dinv, x1, NE, 64);
    k_bias_relu<<<cdiv((long long)NN * 64, B), B, 0, stream>>>(x1, b1, NN, 64);

    // ---- layer 2: K=64 -> F=256 (WMMA, 64x16 tile/wave) ----
    k_gemm_wmma<<<dim3(256 / 64, NN / 64), 128, 0, stream>>>(x1, W2, xw, 64, 256, nullptr, 0);
    k_selfloop<<<cdiv((long long)NN * 256, B), B, 0, stream>>>(xw, dinv, x2, NN, 256);
    k_scatter<<<cdiv((long long)NE * 64, B), B, 0, stream>>>(erow, ecol, xw, dinv, x2, NE, 256);
    k_bias_relu<<<cdiv((long long)NN * 256, B), B, 0, stream>>>(x2, b2, NN, 256);

    // ---- layer 3: K=256 -> F=512 (WMMA) ----
    k_gemm_wmma<<<dim3(512 / 64, NN / 64), 128, 0, stream>>>(x2, W3, xw, 256, 512, nullptr, 0);
    k_selfloop<<<cdiv((long long)NN * 512, B), B, 0, stream>>>(xw, dinv, x3, NN, 512);
    k_scatter<<<cdiv((long long)NE * 128, B), B, 0, stream>>>(erow, ecol, xw, dinv, x3, NE, 512);
    k_bias_relu<<<cdiv((long long)NN * 512, B), B, 0, stream>>>(x3, b3, NN, 512);

    // ---- pooling: [64, 1024] = concat(mean, max) ----
    k_pool<<<NG, 256, 0, stream>>>(x3, h);

    // ---- MLP head: relu(h @ Wm1 + bm1) via WMMA (NG=64 rows = one band) ----
    k_gemm_wmma<<<dim3(512 / 64, NG / 64), 128, 0, stream>>>(h, Wm1, h2, 1024, 512, bm1, 1);
    k_mlp2<<<cdiv(NG * 10, 64), 64, 0, stream>>>(h2, Wm2, bm2, out);
}